// NestedThresholdingAutoEncoderTopK_60301340836392
// MI455X (gfx1250) — compile-verified
//
#include <hip/hip_runtime.h>

// Problem constants (from reference)
#define B_ 4096
#define D_ 2048
#define F_ 32768
#define K_ 256

typedef float v2f __attribute__((ext_vector_type(2)));
typedef float v8f __attribute__((ext_vector_type(8)));
typedef int b128v __attribute__((vector_size(16)));  // matches async-LDS builtin param

// GEMM tiling: 64x128 block tile, K-step 16, 256 threads = 8 waves (wave32),
// wave grid 2(M)x4(N), each wave owns 2x2 16x16 WMMA subtiles.
#define TM 64
#define TN 128
#define TK 16
#define XPAD 20  // pad A-tile row stride to 20 floats (80B, 16B aligned)

#define AS1 __attribute__((address_space(1)))
#define AS3 __attribute__((address_space(3)))

#if defined(__has_builtin)
#if __has_builtin(__builtin_amdgcn_global_load_async_to_lds_b128)
#define HAVE_ASYNC_LD 1
#else
#define HAVE_ASYNC_LD 0
#endif
#if __has_builtin(__builtin_amdgcn_global_store_async_from_lds_b128)
#define HAVE_ASYNC_ST 1
#else
#define HAVE_ASYNC_ST 0
#endif
#if __has_builtin(__builtin_amdgcn_s_wait_asynccnt)
#define HAVE_ASYNC_WAIT 1
#else
#define HAVE_ASYNC_WAIT 0
#endif
#else
#define HAVE_ASYNC_LD 0
#define HAVE_ASYNC_ST 0
#define HAVE_ASYNC_WAIT 0
#endif

// 16B global -> LDS, direct DMA when available (ASYNCcnt-tracked), else via VGPRs.
__device__ __forceinline__ void g2l_16(const float* __restrict__ g, float* l) {
#if HAVE_ASYNC_LD
  __builtin_amdgcn_global_load_async_to_lds_b128((AS1 b128v*)g, (AS3 b128v*)l, 0, 0);
#else
  *(float4*)l = *(const float4*)g;
#endif
}

// 16B LDS -> global, direct DMA when available, else via VGPRs.
__device__ __forceinline__ void l2g_16(float* g, const float* l) {
#if HAVE_ASYNC_ST
  __builtin_amdgcn_global_store_async_from_lds_b128((AS1 b128v*)g, (AS3 b128v*)l, 0, 0);
#else
  *(float4*)g = *(const float4*)l;
#endif
}

__device__ __forceinline__ void wait_async0() {
#if HAVE_ASYNC_LD || HAVE_ASYNC_ST
#if HAVE_ASYNC_WAIT
  __builtin_amdgcn_s_wait_asynccnt(0);
#else
  asm volatile("s_wait_asynccnt 0x0" ::: "memory");
#endif
#endif
}

__device__ __forceinline__ unsigned absbits(float v) {
  return __float_as_uint(v) & 0x7fffffffu;
}

// ---------------------------------------------------------------------------
// Kernel 1: feat = (x - b_dec) @ W      x:(B,D) W:(D,F) feat:(B,F)
// f32 WMMA 16x16x4. A layout per ISA: lane&15 = M, VGPR pair = K{2*half, 2*half+1}.
// ---------------------------------------------------------------------------
__global__ __launch_bounds__(256) void sae_encode(const float* __restrict__ x,
                                                  const float* __restrict__ W,
                                                  const float* __restrict__ bdec,
                                                  float* __restrict__ feat) {
  __shared__ float xs[TM * XPAD];   // A tile: 64 rows x 16 k (padded)
  __shared__ float ws[TK * TN];     // B tile: 16 k x 128 n

  const int tid   = threadIdx.x;
  const int wave  = tid >> 5;
  const int lane  = tid & 31;
  const int lhalf = lane >> 4;
  const int l15   = lane & 15;
  const int mBase = blockIdx.y * TM;
  const int nBase = blockIdx.x * TN;
  const int wm = (wave >> 2) * 32;  // wave's M offset within block tile
  const int wn = (wave & 3) * 32;   // wave's N offset within block tile

  v8f acc[2][2] = {};

  for (int k0 = 0; k0 < D_; k0 += TK) {
    // Stage B first: async DMA straight into LDS (dominant W traffic).
#pragma unroll
    for (int p = 0; p < 2; ++p) {
      const int kk = (tid >> 5) + p * 8;
      const int nq = (tid & 31) * 4;
      g2l_16(W + (size_t)(k0 + kk) * F_ + nBase + nq, &ws[kk * TN + nq]);
    }
    // Stage A via VGPRs (needs the -b_dec transform).
    {
      const int row = tid >> 2, q = tid & 3;
      float4 v  = *(const float4*)(x + (size_t)(mBase + row) * D_ + k0 + q * 4);
      float4 bb = *(const float4*)(bdec + k0 + q * 4);
      v.x -= bb.x; v.y -= bb.y; v.z -= bb.z; v.w -= bb.w;
      *(float4*)(&xs[row * XPAD + q * 4]) = v;
    }
    if (k0 + TK < D_) {
      __builtin_prefetch(W + (size_t)(k0 + TK) * F_ + nBase + (tid & 31) * 4, 0, 1);
    }
    wait_async0();
    __syncthreads();

#pragma unroll
    for (int kk = 0; kk < TK; kk += 4) {
      const int kc = kk + 2 * lhalf;
      v2f a[2], b[2];
#pragma unroll
      for (int mi = 0; mi < 2; ++mi)
        a[mi] = *(const v2f*)(&xs[(wm + mi * 16 + l15) * XPAD + kc]);
#pragma unroll
      for (int ni = 0; ni < 2; ++ni) {
        const int col = wn + ni * 16 + l15;
        v2f t;
        t.x = ws[kc * TN + col];
        t.y = ws[(kc + 1) * TN + col];
        b[ni] = t;
      }
#pragma unroll
      for (int mi = 0; mi < 2; ++mi)
#pragma unroll
        for (int ni = 0; ni < 2; ++ni)
          acc[mi][ni] = __builtin_amdgcn_wmma_f32_16x16x4_f32(
              false, a[mi], false, b[ni], (short)0, acc[mi][ni], false, false);
    }
    __syncthreads();
  }

  // C layout: lane&15 = N, VGPR r -> M = 8*(lane>>4) + r
#pragma unroll
  for (int mi = 0; mi < 2; ++mi)
#pragma unroll
    for (int ni = 0; ni < 2; ++ni) {
      const int col = nBase + wn + ni * 16 + l15;
#pragma unroll
      for (int r = 0; r < 8; ++r) {
        const int rowc = mBase + wm + mi * 16 + lhalf * 8 + r;
        feat[(size_t)rowc * F_ + col] = acc[mi][ni][r];
      }
    }
}

// ---------------------------------------------------------------------------
// Kernel 2: per-row exact top-256 by |value|; zero everything else IN PLACE.
// Whole 128KB row lives in LDS (320KB/WGP on CDNA5). 4-round radix histogram
// on the 31-bit abs-key finds the exact K-th threshold; ties broken by lowest
// index via LDS prefix scan -> deterministic.
// ---------------------------------------------------------------------------
#define TPK 512
__global__ __launch_bounds__(TPK) void sae_topk(float* __restrict__ feat) {
  extern __shared__ float sval[];     // F_ floats (dynamic, 128KB)
  __shared__ int hist[256];
  __shared__ int scanbuf[TPK];
  __shared__ unsigned s_prefix;
  __shared__ int s_need;

  const int tid = threadIdx.x;
  float* grow = feat + (size_t)blockIdx.x * F_;

  // DMA full row into LDS (ASYNCcnt), 16 b128 transfers per thread.
  for (int i = tid * 4; i < F_; i += TPK * 4)
    g2l_16(grow + i, &sval[i]);
  wait_async0();
  __syncthreads();

  // Radix-select the K_-th largest abs key.
  unsigned prefix = 0;
  int need = K_;
  for (int s = 24; s >= 0; s -= 8) {
    for (int i = tid; i < 256; i += TPK) hist[i] = 0;
    __syncthreads();
    for (int i = tid; i < F_; i += TPK) {
      const unsigned u = absbits(sval[i]);
      const bool match = (s == 24) || ((u >> (s + 8)) == (prefix >> (s + 8)));
      if (match) atomicAdd(&hist[(u >> s) & 255], 1);
    }
    __syncthreads();
    if (tid == 0) {
      int cum = 0;
      unsigned g = 0;
      for (int d = 255; d >= 0; --d) {
        const int h = hist[d];
        if (cum + h >= need) { g = (unsigned)d; break; }
        cum += h;
      }
      s_prefix = prefix | (g << s);
      s_need = need - cum;
    }
    __syncthreads();
    prefix = s_prefix;
    need = s_need;
    __syncthreads();
  }
  const unsigned T = prefix;  // exact key of the K-th largest
  const int r = need;         // how many ties (==T) to keep, lowest index first

  // Deterministic tie ranking: contiguous 64-element chunks + block prefix scan.
  const int base = tid * (F_ / TPK);
  int myties = 0;
  for (int i = 0; i < F_ / TPK; ++i)
    if (absbits(sval[base + i]) == T) ++myties;
  scanbuf[tid] = myties;
  __syncthreads();
  for (int off = 1; off < TPK; off <<= 1) {
    int t = 0;
    if (tid >= off) t = scanbuf[tid - off];
    __syncthreads();
    if (t) scanbuf[tid] += t;
    __syncthreads();
  }
  const int tieoff = scanbuf[tid] - myties;  // exclusive scan

  // Apply selection in LDS (thread-private chunk), then bulk DMA back out.
  int trank = 0;
  for (int i = 0; i < F_ / TPK; ++i) {
    const float v = sval[base + i];
    const unsigned u = absbits(v);
    bool sel = false;
    if (u > T) {
      sel = true;
    } else if (u == T) {
      sel = (tieoff + trank) < r;
      ++trank;
    }
    sval[base + i] = sel ? v : 0.0f;
  }
  __syncthreads();
  for (int i = tid * 4; i < F_; i += TPK * 4)
    l2g_16(grow + i, &sval[i]);
  // s_endpgm performs an implicit wait-idle; all async stores drain there.
}

// ---------------------------------------------------------------------------
// Kernel 3: out = enc @ W^T + b_dec    enc:(B,F) W:(D,F) out:(B,D)
// B-matrix B[k][n] = W[n][k]: coalesced reads of W rows, transposed into LDS.
// ---------------------------------------------------------------------------
__global__ __launch_bounds__(256) void sae_decode(const float* __restrict__ enc,
                                                  const float* __restrict__ W,
                                                  const float* __restrict__ bdec,
                                                  float* __restrict__ out) {
  __shared__ float as_[TM * XPAD];  // A tile: 64 rows x 16 k
  __shared__ float bs[TK * TN];     // B tile: 16 k x 128 n (transposed in)

  const int tid   = threadIdx.x;
  const int wave  = tid >> 5;
  const int lane  = tid & 31;
  const int lhalf = lane >> 4;
  const int l15   = lane & 15;
  const int mBase = blockIdx.y * TM;
  const int nBase = blockIdx.x * TN;  // N = activation dim here (2048)
  const int wm = (wave >> 2) * 32;
  const int wn = (wave & 3) * 32;

  v8f acc[2][2] = {};

  for (int k0 = 0; k0 < F_; k0 += TK) {
    // Stage A from enc: async DMA (no transform needed). Note padded LDS rows:
    // XPAD=20 keeps every (row*XPAD + q*4) float4 16B-aligned.
    {
      const int row = tid >> 2, q = tid & 3;
      g2l_16(enc + (size_t)(mBase + row) * F_ + k0 + q * 4,
             &as_[row * XPAD + q * 4]);
    }
    // Stage B transposed: read W[n][k0..k0+15] contiguously, scatter into bs[k][n].
    // LDS store bank = n mod 64 -> the 4 lanes sharing an n (q=0..3) hit one bank.
    for (int idx = tid; idx < TN * 4; idx += 256) {
      const int n = idx >> 2, q = idx & 3;
      const float4 v = *(const float4*)(W + (size_t)(nBase + n) * F_ + k0 + q * 4);
      bs[(q * 4 + 0) * TN + n] = v.x;
      bs[(q * 4 + 1) * TN + n] = v.y;
      bs[(q * 4 + 2) * TN + n] = v.z;
      bs[(q * 4 + 3) * TN + n] = v.w;
    }
    if (k0 + TK < F_) {
      __builtin_prefetch(enc + (size_t)(mBase + (tid >> 2)) * F_ + k0 + TK, 0, 1);
    }
    wait_async0();
    __syncthreads();

#pragma unroll
    for (int kk = 0; kk < TK; kk += 4) {
      const int kc = kk + 2 * lhalf;
      v2f a[2], b[2];
#pragma unroll
      for (int mi = 0; mi < 2; ++mi)
        a[mi] = *(const v2f*)(&as_[(wm + mi * 16 + l15) * XPAD + kc]);
#pragma unroll
      for (int ni = 0; ni < 2; ++ni) {
        const int col = wn + ni * 16 + l15;
        v2f t;
        t.x = bs[kc * TN + col];
        t.y = bs[(kc + 1) * TN + col];
        b[ni] = t;
      }
#pragma unroll
      for (int mi = 0; mi < 2; ++mi)
#pragma unroll
        for (int ni = 0; ni < 2; ++ni)
          acc[mi][ni] = __builtin_amdgcn_wmma_f32_16x16x4_f32(
              false, a[mi], false, b[ni], (short)0, acc[mi][ni], false, false);
    }
    __syncthreads();
  }

#pragma unroll
  for (int mi = 0; mi < 2; ++mi)
#pragma unroll
    for (int ni = 0; ni < 2; ++ni) {
      const int col = nBase + wn + ni * 16 + l15;
      const float bb = bdec[col];
#pragma unroll
      for (int r = 0; r < 8; ++r) {
        const int rowc = mBase + wm + mi * 16 + lhalf * 8 + r;
        out[(size_t)rowc * D_ + col] = acc[mi][ni][r] + bb;
      }
    }
}

// ---------------------------------------------------------------------------
extern "C" void kernel_launch(void* const* d_in, const int* in_sizes, int n_in,
                              void* d_out, int out_size, void* d_ws, size_t ws_size,
                              hipStream_t stream) {
  (void)in_sizes; (void)n_in; (void)out_size; (void)ws_size;
  const float* x    = (const float*)d_in[0];
  const float* W    = (const float*)d_in[1];
  const float* bdec = (const float*)d_in[2];
  float* out  = (float*)d_out;
  float* feat = (float*)d_ws;  // (B_, F_) f32 scratch; becomes enc after top-k

  dim3 eg(F_ / TN, B_ / TM);  // 256 x 64 blocks
  sae_encode<<<eg, 256, 0, stream>>>(x, W, bdec, feat);

  sae_topk<<<B_, TPK, F_ * sizeof(float), stream>>>(feat);

  dim3 dg(D_ / TN, B_ / TM);  // 16 x 64 blocks
  sae_decode<<<dg, 256, 0, stream>>>(feat, W, bdec, out);
}